// SSKernelSlow_52956946760021
// MI455X (gfx1250) — compile-verified
//
#include <hip/hip_runtime.h>
#include <cstddef>

// SSKernelSlow for MI455X (gfx1250): Krylov U/V factorization + Newton-Schulz
// inverse, all heavy math as complex 128x128x128 GEMMs on V_WMMA_F32_16X16X4_F32.
// H=1024 heads, N=128, L=4096. One 256-thread workgroup (8 wave32) per head.
// Workspace requirement: 1024 heads * 131072 floats = 512 MB in d_ws.

#define BSTRIDE 136                       // padded LDS row stride (floats), bank-conflict free
#define BPLANE  (128 * BSTRIDE)           // one staged operand plane
#define BSTAGE_FLOATS (2 * BPLANE)        // re + im planes
#define LDS_FLOATS (BSTAGE_FLOATS + 512)  // + 4 x 128-float work vectors
#define HEAD_WS_FLOATS 131072             // 3 complex NxN mats + V(128x64) + U(64x128)

typedef float v2f __attribute__((ext_vector_type(2)));
typedef float v8f __attribute__((ext_vector_type(8)));

__device__ __forceinline__ v8f wmma4(v2f a, v2f b, v8f c) {
  // D(16x16,f32) = A(16x4,f32) * B(4x16,f32) + C
  return __builtin_amdgcn_wmma_f32_16x16x4_f32(false, a, false, b, (short)0, c,
                                               false, false);
}

// Complex GEMM: acc = A(MxK=128) * B(128 x Nw), Nw = NT*16.
// B (both planes) staged in LDS. Wave w owns row-stripe [16w,16w+16).
// epi: 0: D=acc   1: D=E-acc   2: D=2E-acc   3: D=acc+E
template <int NT>
__device__ void cgemm(const float* Are, const float* Aim, int sA,
                      const float* Bre, const float* Bim, int sB,
                      float* Dre, float* Dim, int sD,
                      const float* Ere, const float* Eim, int sE,
                      int M, int epi, int storeIm, float* lds)
{
  constexpr int Nw = NT * 16;
  const int tid = threadIdx.x;
  float* ldsRe = lds;
  float* ldsIm = lds + BPLANE;
  for (int idx = tid; idx < 128 * Nw; idx += 256) {
    int r = idx / Nw, c = idx - r * Nw;
    ldsRe[r * BSTRIDE + c] = Bre[r * sB + c];
    ldsIm[r * BSTRIDE + c] = Bim[r * sB + c];
  }
  __syncthreads();

  const int wave = tid >> 5, lane = tid & 31;
  const int lhalf = lane >> 4, lmod = lane & 15;
  const int MT = M >> 4;
  if (wave < MT) {
    const int row = wave * 16 + lmod;             // A-fragment: lane = M row
    v8f accR[NT], accI[NT];
#pragma unroll
    for (int c = 0; c < NT; ++c) {
      accR[c] = (v8f){0.f,0.f,0.f,0.f,0.f,0.f,0.f,0.f};
      accI[c] = (v8f){0.f,0.f,0.f,0.f,0.f,0.f,0.f,0.f};
    }
    for (int k0 = 0; k0 < 128; k0 += 4) {
      const int ka = k0 + 2 * lhalf;              // A: vgpr v <-> K = ka + v
      v2f ar = *(const v2f*)(Are + row * sA + ka);
      v2f ai = *(const v2f*)(Aim + row * sA + ka);
      const int rb = ka * BSTRIDE + lmod;         // B: vgpr v <-> K row ka+v, lane = N col
#pragma unroll
      for (int c = 0; c < NT; ++c) {
        const int cb = rb + c * 16;
        v2f br, bi, nbi;
        br.x = ldsRe[cb]; br.y = ldsRe[cb + BSTRIDE];
        bi.x = ldsIm[cb]; bi.y = ldsIm[cb + BSTRIDE];
        nbi.x = -bi.x; nbi.y = -bi.y;
        accR[c] = wmma4(ar, br,  accR[c]);        // Re += Ar*Br
        accR[c] = wmma4(ai, nbi, accR[c]);        // Re -= Ai*Bi
        accI[c] = wmma4(ar, bi,  accI[c]);        // Im += Ar*Bi
        accI[c] = wmma4(ai, br,  accI[c]);        // Im += Ai*Br
      }
    }
#pragma unroll
    for (int c = 0; c < NT; ++c) {
#pragma unroll
      for (int j = 0; j < 8; ++j) {               // C/D: vgpr j <-> M = j + 8*lhalf
        const int rr = wave * 16 + j + 8 * lhalf;
        const int cc = c * 16 + lmod;
        float vr = accR[c][j], vi = accI[c][j];
        if (epi == 1)      { vr = Ere[rr*sE+cc] - vr;        vi = Eim[rr*sE+cc] - vi; }
        else if (epi == 2) { vr = 2.f*Ere[rr*sE+cc] - vr;    vi = 2.f*Eim[rr*sE+cc] - vi; }
        else if (epi == 3) { vr += Ere[rr*sE+cc];            vi += Eim[rr*sE+cc]; }
        Dre[rr * sD + cc] = vr;
        if (storeIm) Dim[rr * sD + cc] = vi;
      }
    }
  }
  __threadfence();
  __syncthreads();
}

__global__ void __launch_bounds__(256, 1)
ssm_krylov_kernel(const float* __restrict__ Ain, const float* __restrict__ Bin,
                  const float* __restrict__ Cin, const float* __restrict__ logdt,
                  float* __restrict__ out, float* __restrict__ ws)
{
  extern __shared__ float lds[];
  const int h = blockIdx.x;
  const int tid = threadIdx.x;

  float* slab = ws + (size_t)h * HEAD_WS_FLOATS;
  float *Xre = slab,          *Xim = slab + 16384;   // b0
  float *Yre = slab + 32768,  *Yim = slab + 49152;   // b1
  float *Tre = slab + 65536,  *Tim = slab + 81920;   // b2
  float *Vre = slab + 98304,  *Vim = slab + 106496;  // 128x64 each (stride 64)
  float *Ure = slab + 114688, *Uim = slab + 122880;  // 64x128 each (stride 128)

  const float dt = expf(logdt[h]);
  const float s  = 0.5f * dt;

  // ---- Phase 0: X = (dt/2)*A (interleaved -> planar), Y0 = I + X -------------
  const float* Ah = Ain + (size_t)h * 32768;
  for (int idx = tid; idx < 16384; idx += 256) {
    float re = Ah[2 * idx], im = Ah[2 * idx + 1];
    int r = idx >> 7, c = idx & 127;
    float xr = s * re, xi = s * im;
    Xre[idx] = xr; Xim[idx] = xi;
    Yre[idx] = xr + (r == c ? 1.f : 0.f);
    Yim[idx] = xi;
  }
  __threadfence(); __syncthreads();

  // ---- Phase 1: Newton-Schulz, 4 iters: M = (I - X)^-1 ----------------------
  for (int it = 0; it < 4; ++it) {
    // T = Y - X*Y
    cgemm<8>(Xre,Xim,128, Yre,Yim,128, Tre,Tim,128, Yre,Yim,128, 128, 1, 1, lds);
    // Y = 2Y - Y*T   (in-place; each wave owns its 16-row stripe)
    cgemm<8>(Yre,Yim,128, Tre,Tim,128, Yre,Yim,128, Yre,Yim,128, 128, 2, 1, lds);
  }

  // ---- Phase 2: dA = M*X + M  (into b0; X was staged to LDS first) ----------
  cgemm<8>(Yre,Yim,128, Xre,Xim,128, Xre,Xim,128, Yre,Yim,128, 128, 3, 1, lds);
  float *dAre = Xre, *dAim = Xim;

  // ---- Phase 3: dB = dt * M * B -> V[:,0] and LDS vector --------------------
  float* xr = lds + BSTAGE_FLOATS;  // current vector (re)
  float* xi = xr + 128;
  float* nr = xi + 128;             // next vector (re)
  float* ni = nr + 128;
  const float* Bh = Bin + (size_t)h * 256;
  if (tid < 128) { nr[tid] = Bh[2 * tid]; ni[tid] = Bh[2 * tid + 1]; }
  __syncthreads();
  if (tid < 128) {
    float ar = 0.f, ai = 0.f;
    const float* mr = Yre + tid * 128;
    const float* mi = Yim + tid * 128;
    for (int n = 0; n < 128; ++n) {
      float br = nr[n], bi = ni[n];
      ar += mr[n] * br - mi[n] * bi;
      ai += mr[n] * bi + mi[n] * br;
    }
    ar *= dt; ai *= dt;
    xr[tid] = ar; xi[tid] = ai;
    Vre[tid * 64] = ar; Vim[tid * 64] = ai;
  }
  __syncthreads();

  // ---- Phase 4: v_i = dA * v_{i-1}, i = 1..15 (VALU mat-vecs) ---------------
  for (int i = 1; i < 16; ++i) {
    if (tid < 128) {
      float ar = 0.f, ai = 0.f;
      const float* mr = dAre + tid * 128;
      const float* mi = dAim + tid * 128;
      for (int n = 0; n < 128; ++n) {
        float br = xr[n], bi = xi[n];
        ar += mr[n] * br - mi[n] * bi;
        ai += mr[n] * bi + mi[n] * br;
      }
      nr[tid] = ar; ni[tid] = ai;
      Vre[tid * 64 + i] = ar; Vim[tid * 64 + i] = ai;
    }
    __syncthreads();
    if (tid < 128) { xr[tid] = nr[tid]; xi[tid] = ni[tid]; }
    __syncthreads();
  }
  __threadfence(); __syncthreads();

  // ---- Phase 5-9: V doubling via dA^16, dA^32; P = dA^64 --------------------
  cgemm<8>(dAre,dAim,128, dAre,dAim,128, Yre,Yim,128, nullptr,nullptr,0, 128,0,1, lds); // W2 -> Y
  cgemm<8>(Yre,Yim,128,  Yre,Yim,128,  Tre,Tim,128, nullptr,nullptr,0, 128,0,1, lds);   // W4 -> T
  cgemm<8>(Tre,Tim,128,  Tre,Tim,128,  Yre,Yim,128, nullptr,nullptr,0, 128,0,1, lds);   // W8 -> Y
  cgemm<8>(Yre,Yim,128,  Yre,Yim,128,  Tre,Tim,128, nullptr,nullptr,0, 128,0,1, lds);   // W16 -> T
  cgemm<1>(Tre,Tim,128,  Vre,Vim,64,   Vre+16,Vim+16,64, nullptr,nullptr,0, 128,0,1, lds); // V[:,16:32]
  cgemm<8>(Tre,Tim,128,  Tre,Tim,128,  Yre,Yim,128, nullptr,nullptr,0, 128,0,1, lds);   // W32 -> Y
  cgemm<2>(Yre,Yim,128,  Vre,Vim,64,   Vre+32,Vim+32,64, nullptr,nullptr,0, 128,0,1, lds); // V[:,32:64]
  cgemm<8>(Yre,Yim,128,  Yre,Yim,128,  Tre,Tim,128, nullptr,nullptr,0, 128,0,1, lds);   // P = dA^64 -> T
  float *Pre = Tre, *Pim = Tim;

  // ---- Phase 10: u_j = u_{j-1} * P, j = 0..15 (row mat-vecs, coalesced) -----
  const float* Ch = Cin + (size_t)h * 256;
  if (tid < 128) {
    float cr = Ch[2 * tid], ci = Ch[2 * tid + 1];
    xr[tid] = cr; xi[tid] = ci;
    Ure[tid] = cr; Uim[tid] = ci;
  }
  __syncthreads();
  for (int j = 1; j < 16; ++j) {
    if (tid < 128) {
      float ar = 0.f, ai = 0.f;
      for (int m = 0; m < 128; ++m) {
        float pr = Pre[m * 128 + tid], pi = Pim[m * 128 + tid];
        float ur = xr[m], ui = xi[m];
        ar += ur * pr - ui * pi;
        ai += ur * pi + ui * pr;
      }
      nr[tid] = ar; ni[tid] = ai;
      Ure[j * 128 + tid] = ar; Uim[j * 128 + tid] = ai;
    }
    __syncthreads();
    if (tid < 128) { xr[tid] = nr[tid]; xi[tid] = ni[tid]; }
    __syncthreads();
  }
  __threadfence(); __syncthreads();

  // ---- Phase 11-14: U doubling via P^16 = dA^1024, P^32 = dA^2048 -----------
  cgemm<8>(Pre,Pim,128, Pre,Pim,128, Xre,Xim,128, nullptr,nullptr,0, 128,0,1, lds); // Q2  -> X (dA dead)
  cgemm<8>(Xre,Xim,128, Xre,Xim,128, Yre,Yim,128, nullptr,nullptr,0, 128,0,1, lds); // Q4  -> Y
  cgemm<8>(Yre,Yim,128, Yre,Yim,128, Xre,Xim,128, nullptr,nullptr,0, 128,0,1, lds); // Q8  -> X
  cgemm<8>(Xre,Xim,128, Xre,Xim,128, Yre,Yim,128, nullptr,nullptr,0, 128,0,1, lds); // P16 -> Y
  cgemm<8>(Ure,Uim,128, Yre,Yim,128, Ure+16*128,Uim+16*128,128,
           nullptr,nullptr,0, 16, 0, 1, lds);                                       // U[16:32]
  cgemm<8>(Yre,Yim,128, Yre,Yim,128, Xre,Xim,128, nullptr,nullptr,0, 128,0,1, lds); // P32 -> X
  cgemm<8>(Ure,Uim,128, Xre,Xim,128, Ure+32*128,Uim+32*128,128,
           nullptr,nullptr,0, 32, 0, 1, lds);                                       // U[32:64]

  // ---- Phase 15: out[h, 64j + i] = Re( (U*V)[j][i] ) ------------------------
  cgemm<4>(Ure,Uim,128, Vre,Vim,64, out + (size_t)h * 4096, nullptr, 64,
           nullptr,nullptr,0, 64, 0, 0, lds);
}

extern "C" void kernel_launch(void* const* d_in, const int* in_sizes, int n_in,
                              void* d_out, int out_size, void* d_ws, size_t ws_size,
                              hipStream_t stream) {
  const float* A     = (const float*)d_in[0];  // (H,128,128,2) f32
  const float* B     = (const float*)d_in[1];  // (H,128,2)
  const float* C     = (const float*)d_in[2];  // (1,H,128,2)
  const float* logdt = (const float*)d_in[3];  // (H,)
  float* out = (float*)d_out;                  // (1,H,4096) f32
  float* ws  = (float*)d_ws;                   // needs >= H * 131072 * 4 bytes

  const int H = in_sizes[3];
  const size_t shmem = (size_t)LDS_FLOATS * sizeof(float);  // ~138 KB dynamic LDS
  hipFuncSetAttribute((const void*)ssm_krylov_kernel,
                      hipFuncAttributeMaxDynamicSharedMemorySize, (int)shmem);
  ssm_krylov_kernel<<<dim3(H), dim3(256), shmem, stream>>>(A, B, C, logdt, out, ws);
}